// GAT_71725953843361
// MI455X (gfx1250) — compile-verified
//
#include <hip/hip_runtime.h>
#include <hip/hip_bf16.h>

#define N_NODES 100000
#define N_EDGES 800000
#define NH 12
#define NF 8
#define ND 96
#define SLOPE 0.2f
#define LN_EPS 1e-6f

typedef float v2f __attribute__((ext_vector_type(2)));
typedef float v8f __attribute__((ext_vector_type(8)));

__device__ __forceinline__ void atomicMaxF(float* addr, float val) {
    unsigned int* ua = (unsigned int*)addr;
    unsigned int old = __hip_atomic_load(ua, __ATOMIC_RELAXED, __HIP_MEMORY_SCOPE_AGENT);
    while (true) {
        float of = __uint_as_float(old);
        if (of >= val) break;                   // -inf sentinel compares false for any larger val -> replaced
        unsigned int assumed = old;
        old = atomicCAS(ua, assumed, __float_as_uint(val));
        if (old == assumed) break;
    }
}

// ---- K0: set global-max scalar to -inf (denom/agg zeroed by hipMemsetAsync) ----
__global__ void k_set_gmax(float* gmax) {
    if (threadIdx.x == 0) *gmax = __uint_as_float(0xff800000u); // -inf
}

// ---- K1: per-(node,head) score dot products ----
__global__ void k_node_scores(const float* __restrict__ x,
                              const float* __restrict__ wsrc,
                              const float* __restrict__ wtrg,
                              float* __restrict__ s_src,
                              float* __restrict__ s_trg) {
    int i = blockIdx.x * blockDim.x + threadIdx.x;   // over N*NH
    if (i >= N_NODES * NH) return;
    int n = i / NH, h = i - n * NH;
    const float* p = x + n * ND + h * NF;
    const float* a = wsrc + h * NF;
    const float* b = wtrg + h * NF;
    float sa = 0.f, sb = 0.f;
#pragma unroll
    for (int f = 0; f < NF; ++f) { float v = p[f]; sa += v * a[f]; sb += v * b[f]; }
    s_src[i] = sa;
    s_trg[i] = sb;
}

// ---- K2: global max of leaky-relu'd edge scores ----
__global__ void k_gmax(const int* __restrict__ ei,
                       const float* __restrict__ s_src,
                       const float* __restrict__ s_trg,
                       float* gmax) {
    int e = blockIdx.x * blockDim.x + threadIdx.x;
    float m = __uint_as_float(0xff800000u);
    if (e < N_EDGES) {
        int s = ei[e], t = ei[N_EDGES + e];
#pragma unroll
        for (int h = 0; h < NH; ++h) {
            float sc = s_src[s * NH + h] + s_trg[t * NH + h];
            sc = (sc >= 0.f) ? sc : SLOPE * sc;
            m = fmaxf(m, sc);
        }
    }
#pragma unroll
    for (int off = 16; off > 0; off >>= 1) m = fmaxf(m, __shfl_xor(m, off, 32));
    if ((threadIdx.x & 31) == 0) atomicMaxF(gmax, m);
}

// ---- K3: segment-softmax denominators ----
__global__ void k_denom(const int* __restrict__ ei,
                        const float* __restrict__ s_src,
                        const float* __restrict__ s_trg,
                        const float* __restrict__ gmax,
                        float* __restrict__ denom) {
    int e = blockIdx.x * blockDim.x + threadIdx.x;
    if (e >= N_EDGES) return;
    int s = ei[e], t = ei[N_EDGES + e];
    float m = *gmax;
#pragma unroll
    for (int h = 0; h < NH; ++h) {
        float sc = s_src[s * NH + h] + s_trg[t * NH + h];
        sc = (sc >= 0.f) ? sc : SLOPE * sc;
        atomicAdd(&denom[t * NH + h], __expf(sc - m));
    }
}

// ---- K4: weighted scatter-aggregation (one thread per edge-head) ----
__global__ void k_agg(const int* __restrict__ ei,
                      const float* __restrict__ x,
                      const float* __restrict__ s_src,
                      const float* __restrict__ s_trg,
                      const float* __restrict__ gmax,
                      const float* __restrict__ denom,
                      float* __restrict__ agg) {
    int idx = blockIdx.x * blockDim.x + threadIdx.x;   // E*NH = 9.6M < 2^31
    if (idx >= N_EDGES * NH) return;
    int e = idx / NH, h = idx - e * NH;
    int s = ei[e], t = ei[N_EDGES + e];
    float sc = s_src[s * NH + h] + s_trg[t * NH + h];
    sc = (sc >= 0.f) ? sc : SLOPE * sc;
    float attn = __expf(sc - *gmax) / (denom[t * NH + h] + 1e-16f);
    const float* ps = x + s * ND + h * NF;
    float* pa = agg + t * ND + h * NF;
#pragma unroll
    for (int f = 0; f < NF; ++f) atomicAdd(&pa[f], ps[f] * attn);
}

// ---- K5: RMS norm in place, one wave32 per node ----
__global__ void k_rms(const float* __restrict__ ln_w, float* __restrict__ agg) {
    int warp = (blockIdx.x * blockDim.x + threadIdx.x) >> 5;
    int lane = threadIdx.x & 31;
    if (warp >= N_NODES) return;
    float* row = agg + warp * ND;
    float v0 = row[lane], v1 = row[lane + 32], v2 = row[lane + 64];
    float ss = v0 * v0 + v1 * v1 + v2 * v2;
#pragma unroll
    for (int off = 16; off > 0; off >>= 1) ss += __shfl_xor(ss, off, 32);
    float inv = rsqrtf(ss * (1.0f / ND) + LN_EPS);
    row[lane]      = v0 * inv * ln_w[lane];
    row[lane + 32] = v1 * inv * ln_w[lane + 32];
    row[lane + 64] = v2 * inv * ln_w[lane + 64];
}

// ---- K6: out = x + y @ w^T via V_WMMA_F32_16X16X4_F32, one wave per 16x16 tile ----
// A(16x4): lanes 0-15 -> M=lane, K={k0,k0+1}; lanes 16-31 -> M=lane-16, K={k0+2,k0+3}
// B(4x16): lanes 0-15 -> N=lane, K={k0,k0+1}; lanes 16-31 -> N=lane-16, K={k0+2,k0+3}; B[k][n]=w[n][k]
// C/D(16x16): VGPR v -> M=v (+8 for lanes 16-31), N=lane&15
__global__ void k_matmul(const float* __restrict__ y,
                         const float* __restrict__ w,
                         const float* __restrict__ x,
                         float* __restrict__ out) {
    int wid  = (blockIdx.x * blockDim.x + threadIdx.x) >> 5;  // 0..37499, grid exact
    int lane = threadIdx.x & 31;
    int nt = wid / (ND / 16);       // node tile 0..6249
    int dt = wid - nt * (ND / 16);  // out-col tile 0..5
    int col = lane & 15;
    int hi  = lane >> 4;            // 0 | 1
    int kb  = hi * 2;
    const float* yb = y + (nt * 16 + col) * ND;
    const float* wb = w + (dt * 16 + col) * ND;
    v8f c = {};
#pragma unroll
    for (int k0 = 0; k0 < ND; k0 += 4) {
        v2f a, b;
        a.x = yb[k0 + kb]; a.y = yb[k0 + kb + 1];
        b.x = wb[k0 + kb]; b.y = wb[k0 + kb + 1];
        c = __builtin_amdgcn_wmma_f32_16x16x4_f32(false, a, false, b, (short)0, c, false, false);
    }
#pragma unroll
    for (int v = 0; v < 8; ++v) {
        int n = nt * 16 + v + hi * 8;
        int d = dt * 16 + col;
        out[n * ND + d] = x[n * ND + d] + c[v];
    }
}

extern "C" void kernel_launch(void* const* d_in, const int* in_sizes, int n_in,
                              void* d_out, int out_size, void* d_ws, size_t ws_size,
                              hipStream_t stream) {
    const float* x      = (const float*)d_in[0];
    const int*   ei     = (const int*)d_in[1];
    const float* w_proj = (const float*)d_in[2];
    const float* wsrc   = (const float*)d_in[3];
    const float* wtrg   = (const float*)d_in[4];
    const float* ln_w   = (const float*)d_in[5];
    float* out = (float*)d_out;

    // workspace layout (floats)
    float* ws    = (float*)d_ws;
    float* gmax  = ws;                       // 1 (padded to 64)
    float* s_src = ws + 64;                  // N*NH
    float* s_trg = s_src + N_NODES * NH;     // N*NH
    float* denom = s_trg + N_NODES * NH;     // N*NH
    float* agg   = denom + N_NODES * NH;     // N*ND  (reused as y after rmsnorm)

    // zero denom + agg (contiguous), set gmax = -inf
    hipMemsetAsync(denom, 0, (size_t)(N_NODES * NH + N_NODES * ND) * sizeof(float), stream);
    k_set_gmax<<<1, 32, 0, stream>>>(gmax);

    k_node_scores<<<(N_NODES * NH + 255) / 256, 256, 0, stream>>>(x, wsrc, wtrg, s_src, s_trg);
    k_gmax<<<(N_EDGES + 255) / 256, 256, 0, stream>>>(ei, s_src, s_trg, gmax);
    k_denom<<<(N_EDGES + 255) / 256, 256, 0, stream>>>(ei, s_src, s_trg, gmax, denom);
    k_agg<<<(N_EDGES * NH + 255) / 256, 256, 0, stream>>>(ei, x, s_src, s_trg, gmax, denom, agg);
    k_rms<<<(N_NODES * 32 + 255) / 256, 256, 0, stream>>>(ln_w, agg);

    // 6250 node-tiles * 6 col-tiles = 37500 waves; 4 waves/block -> 9375 blocks (exact)
    k_matmul<<<(N_NODES / 16) * (ND / 16) / 4, 128, 0, stream>>>(agg, w_proj, x, out);
}